// MeshTokenizer_25606595019046
// MI455X (gfx1250) — compile-verified
//
#include <hip/hip_runtime.h>
#include <hip/hip_bf16.h>
#include <stdint.h>

#define B_BATCH 16
#define TLOC 8192         // elements per LDS sort tile (8192 * 16B = 128 KB LDS; CDNA5 allows 320 KB/WG)
#define LBS 512           // threads per block for local sort kernels (16 waves, wave32)

#ifndef USE_ASYNC_LDS
#define USE_ASYNC_LDS 1   // gfx1250 async global<->LDS path (ASYNCcnt)
#endif

// ---- order-preserving float <-> uint32 key (bijective) --------------------
__device__ __forceinline__ unsigned enc_f(float f) {
  unsigned u = __float_as_uint(f);
  return (u & 0x80000000u) ? ~u : (u | 0x80000000u);
}
__device__ __forceinline__ float dec_f(unsigned k) {
  unsigned u = (k & 0x80000000u) ? (k & 0x7fffffffu) : ~k;
  return __uint_as_float(u);
}

// full key compare: (z,y,x) then index (stable lexsort equivalence)
__device__ __forceinline__ bool key_less(const uint4 a, const uint4 b) {
  if (a.x != b.x) return a.x < b.x;
  if (a.y != b.y) return a.y < b.y;
  if (a.z != b.z) return a.z < b.z;
  return a.w < b.w;
}
// coord-only compare (z,y,x) for the within-face sort
__device__ __forceinline__ bool coord_less(const uint4 a, const uint4 b) {
  if (a.x != b.x) return a.x < b.x;
  if (a.y != b.y) return a.y < b.y;
  return a.z < b.z;
}

__device__ __forceinline__ float discretize_f(float t) {
  float s = (t + 1.0f) / 2.0f * 128.0f - 0.5f;  // (t - lo)/(hi-lo)*N - 0.5
  float r = rintf(s);                            // round half-to-even (matches jnp.round)
  return fminf(fmaxf(r, 0.0f), 127.0f);
}

// ---------------------------------------------------------------------------
__global__ void k_init(unsigned* longest_bits) { *longest_bits = 0u; }

// min/max over batch dim per (vertex, coord); center + global longest axis
__global__ void k_minmax(const float* __restrict__ verts, float* __restrict__ center,
                         unsigned* __restrict__ longest_bits, int NV) {
  int idx = blockIdx.x * blockDim.x + threadIdx.x;  // over NV*3
  int total = NV * 3;
  float range = 0.0f;
  if (idx < total) {
    float mn = verts[idx], mx = mn;
    for (int b = 1; b < B_BATCH; ++b) {
      float v = verts[(size_t)b * total + idx];
      mn = fminf(mn, v);
      mx = fmaxf(mx, v);
    }
    center[idx] = (mn + mx) * 0.5f;
    range = mx - mn;
  }
  __shared__ float red[LBS];
  red[threadIdx.x] = range;
  __syncthreads();
  for (int s = LBS / 2; s > 0; s >>= 1) {
    if (threadIdx.x < s) red[threadIdx.x] = fmaxf(red[threadIdx.x], red[threadIdx.x + s]);
    __syncthreads();
  }
  if (threadIdx.x == 0) atomicMax(longest_bits, __float_as_uint(red[0]));  // range >= 0
}

// build per-vertex sort keys from normalized coords (keys decode bijectively
// back to the normalized floats, so no separate vnorm array is needed)
__global__ void k_vkeys(const float* __restrict__ verts, const float* __restrict__ center,
                        const unsigned* __restrict__ longest_bits,
                        uint4* __restrict__ vkeys, int NV) {
  int t = blockIdx.x * blockDim.x + threadIdx.x;  // over B*NV
  if (t >= B_BATCH * NV) return;
  int b = t / NV, i = t % NV;
  float L = __uint_as_float(*longest_bits);
  size_t base = ((size_t)b * NV + i) * 3;
  float x = (verts[base + 0] - center[i * 3 + 0]) / L;
  float y = (verts[base + 1] - center[i * 3 + 1]) / L;
  float z = (verts[base + 2] - center[i * 3 + 2]) / L;
  vkeys[t] = make_uint4(enc_f(z), enc_f(y), enc_f(x), (unsigned)i);
}

// ---- bitonic sort: LDS tile helpers ---------------------------------------
__device__ __forceinline__ void load_tile_async(uint4* __restrict__ gsrc,
                                                uint4* __restrict__ tile) {
#if USE_ASYNC_LDS
  for (int r = threadIdx.x; r < TLOC; r += LBS) {
    unsigned lds_addr = (unsigned)(uintptr_t)(void*)&tile[r];
    unsigned long long ga = (unsigned long long)(uintptr_t)(void*)&gsrc[r];
    // CDNA5 async DMA global -> LDS (per-lane), tracked by ASYNCcnt
    asm volatile("global_load_async_to_lds_b128 %0, %1, off"
                 :: "v"(lds_addr), "v"(ga) : "memory");
  }
  asm volatile("s_wait_asynccnt 0" ::: "memory");
#else
  for (int r = threadIdx.x; r < TLOC; r += LBS) tile[r] = gsrc[r];
#endif
  __syncthreads();
}

// write a fully-synced LDS tile back to global (async DMA LDS -> global)
__device__ __forceinline__ void store_tile_async(uint4* __restrict__ gdst,
                                                 uint4* __restrict__ tile) {
#if USE_ASYNC_LDS
  // Preceding __syncthreads (end of last sort stage) guarantees all DS writes
  // are complete in LDS before the async engine reads them.
  for (int r = threadIdx.x; r < TLOC; r += LBS) {
    unsigned lds_addr = (unsigned)(uintptr_t)(void*)&tile[r];
    unsigned long long ga = (unsigned long long)(uintptr_t)(void*)&gdst[r];
    asm volatile("global_store_async_from_lds_b128 %0, %1, off"
                 :: "v"(ga), "v"(lds_addr) : "memory");
  }
  asm volatile("s_wait_asynccnt 0" ::: "memory");
#else
  for (int r = threadIdx.x; r < TLOC; r += LBS) gdst[r] = tile[r];
#endif
}

// full bitonic sort of one TLOC tile (stages k=2..TLOC entirely in LDS)
__global__ __launch_bounds__(LBS) void k_bitonic_local_sort(uint4* __restrict__ keys, int n) {
  __shared__ uint4 tile[TLOC];
  int tilesPerBatch = n / TLOC;
  int b = blockIdx.x / tilesPerBatch;
  int t = blockIdx.x % tilesPerBatch;
  size_t gbase = (size_t)b * n + (size_t)t * TLOC;
  int ibase = t * TLOC;  // element index within the batch (for direction bit)
  load_tile_async(keys + gbase, tile);
  for (int k = 2; k <= TLOC; k <<= 1) {
    for (int j = k >> 1; j >= 1; j >>= 1) {
      for (int p = threadIdx.x; p < TLOC / 2; p += LBS) {
        int i = (p / j) * (2 * j) + (p % j);
        int q = i + j;
        bool up = (((ibase + i) & k) == 0);
        uint4 A = tile[i], Bv = tile[q];
        if (key_less(Bv, A) == up) { tile[i] = Bv; tile[q] = A; }
      }
      __syncthreads();
    }
  }
  store_tile_async(keys + gbase, tile);
}

// merge stages j = TLOC/2 .. 1 (all intra-tile) for a given global k
__global__ __launch_bounds__(LBS) void k_bitonic_local_merge(uint4* __restrict__ keys,
                                                             int n, int k) {
  __shared__ uint4 tile[TLOC];
  int tilesPerBatch = n / TLOC;
  int b = blockIdx.x / tilesPerBatch;
  int t = blockIdx.x % tilesPerBatch;
  size_t gbase = (size_t)b * n + (size_t)t * TLOC;
  int ibase = t * TLOC;
  load_tile_async(keys + gbase, tile);
  for (int j = TLOC >> 1; j >= 1; j >>= 1) {
    for (int p = threadIdx.x; p < TLOC / 2; p += LBS) {
      int i = (p / j) * (2 * j) + (p % j);
      int q = i + j;
      bool up = (((ibase + i) & k) == 0);
      uint4 A = tile[i], Bv = tile[q];
      if (key_less(Bv, A) == up) { tile[i] = Bv; tile[q] = A; }
    }
    __syncthreads();
  }
  store_tile_async(keys + gbase, tile);
}

// one global compare-exchange pass (stride j >= TLOC)
__global__ void k_bitonic_global(uint4* __restrict__ keys, int n, int k, int j) {
  int tid = blockIdx.x * blockDim.x + threadIdx.x;  // over B * n/2
  int half = n >> 1;
  if (tid >= B_BATCH * half) return;
  int b = tid / half, p = tid % half;
  int i = (p / j) * (2 * j) + (p % j);
  int q = i + j;
  uint4* base = keys + (size_t)b * n;
  bool up = ((i & k) == 0);
  uint4 A = base[i], Bv = base[q];
  if (key_less(Bv, A) == up) { base[i] = Bv; base[q] = A; }
}

// per-face centroid sort keys (centroid from ORIGINAL within-face order)
__global__ void k_fkeys(const int* __restrict__ faces, const uint4* __restrict__ vkeys,
                        uint4* __restrict__ fkeys, int NV, int NF) {
  int t = blockIdx.x * blockDim.x + threadIdx.x;  // over B*NF
  if (t >= B_BATCH * NF) return;
  int b = t / NF, f = t % NF;
  size_t fb = (size_t)t * 3;
  float X[3], Y[3], Z[3];
  for (int m = 0; m < 3; ++m) {
    int vi = faces[fb + m];
    uint4 K = vkeys[(size_t)b * NV + vi];  // vi indexes the SORTED vertex array
    Z[m] = dec_f(K.x); Y[m] = dec_f(K.y); X[m] = dec_f(K.z);
  }
  float cx = ((X[0] + X[1]) + X[2]) / 3.0f;
  float cy = ((Y[0] + Y[1]) + Y[2]) / 3.0f;
  float cz = ((Z[0] + Z[1]) + Z[2]) / 3.0f;
  fkeys[t] = make_uint4(enc_f(cz), enc_f(cy), enc_f(cx), (unsigned)f);
}

// final emit: within-face stable 3-sort, discretize, write all 4 outputs
__global__ void k_emit(const int* __restrict__ faces, const uint4* __restrict__ vkeys,
                       const uint4* __restrict__ fkeys, float* __restrict__ out,
                       int NV, int NF) {
  int t = blockIdx.x * blockDim.x + threadIdx.x;  // over B*NF (sorted positions)
  if (t >= B_BATCH * NF) return;
  int b = t / NF, j = t % NF;
  unsigned f = fkeys[t].w;
  size_t fb = ((size_t)b * NF + f) * 3;
  uint4 K[3];
  for (int m = 0; m < 3; ++m) {
    int vi = faces[fb + m];
    K[m] = vkeys[(size_t)b * NV + vi];
  }
  // stable ranks by (z,y,x)
  bool l01 = coord_less(K[0], K[1]), l10 = coord_less(K[1], K[0]);
  bool l02 = coord_less(K[0], K[2]), l20 = coord_less(K[2], K[0]);
  bool l12 = coord_less(K[1], K[2]), l21 = coord_less(K[2], K[1]);
  bool e01 = !l01 && !l10, e02 = !l02 && !l20, e12 = !l12 && !l21;
  int r0 = (l10 ? 1 : 0) + (l20 ? 1 : 0);
  int r1 = (l01 ? 1 : 0) + (l21 ? 1 : 0) + (e01 ? 1 : 0);
  int r2 = (l02 ? 1 : 0) + (l12 ? 1 : 0) + (e02 ? 1 : 0) + (e12 ? 1 : 0);
  int ord[3];
  ord[r0] = 0; ord[r1] = 1; ord[r2] = 2;

  float vals[9];
  for (int m = 0; m < 3; ++m) {
    uint4 Km = K[ord[m]];
    vals[m * 3 + 0] = discretize_f(dec_f(Km.z));  // x
    vals[m * 3 + 1] = discretize_f(dec_f(Km.y));  // y
    vals[m * 3 + 2] = discretize_f(dec_f(Km.x));  // z
  }

  size_t row = (size_t)9 * NF + 2;
  size_t ids_base = (size_t)b * row + 1 + (size_t)j * 9;
  size_t am_base  = (size_t)B_BATCH * row + ids_base;
  size_t dp_base  = (size_t)2 * B_BATCH * row + ((size_t)b * NF + j) * 9;
  size_t dfc_base = dp_base + (size_t)B_BATCH * NF * 9;
  for (int q = 0; q < 9; ++q) {
    out[ids_base + q] = vals[q];   // input_ids (never PAD: mask all-true here)
    out[am_base + q]  = 1.0f;      // attention_mask interior
    out[dp_base + q]  = vals[q];   // discrete_padded
    out[dfc_base + q] = vals[q];   // discrete_face_coords
  }
}

// placeholder columns (-1) at both ends of input_ids and attention_mask
__global__ void k_edges(float* __restrict__ out, int NF) {
  int t = threadIdx.x;  // 64 = 16 batches * 4 writes
  if (t >= 4 * B_BATCH) return;
  int b = t >> 2, w = t & 3;
  size_t row = (size_t)9 * NF + 2;
  size_t pos = (w & 1) ? (row - 1) : 0;
  size_t base = ((w >> 1) ? (size_t)B_BATCH * row : (size_t)0) + (size_t)b * row + pos;
  out[base] = -1.0f;
}

// ---------------------------------------------------------------------------
extern "C" void kernel_launch(void* const* d_in, const int* in_sizes, int n_in,
                              void* d_out, int out_size, void* d_ws, size_t ws_size,
                              hipStream_t stream) {
  const float* verts = (const float*)d_in[0];
  const int*   faces = (const int*)d_in[1];
  float* out = (float*)d_out;
  int NV = in_sizes[0] / (B_BATCH * 3);  // 65536
  int NF = in_sizes[1] / (B_BATCH * 3);  // 131072

  char* ws = (char*)d_ws;
  float*    center  = (float*)ws;                                 // NV*3 floats
  unsigned* longest = (unsigned*)(ws + (size_t)NV * 3 * 4);       // 16B slot
  uint4*    vkeys   = (uint4*)(ws + (size_t)NV * 3 * 4 + 16);     // B*NV * 16B
  uint4*    fkeys   = (uint4*)((char*)vkeys + (size_t)B_BATCH * NV * 16);  // B*NF * 16B

  k_init<<<1, 1, 0, stream>>>(longest);
  {
    int total = NV * 3;
    k_minmax<<<(total + LBS - 1) / LBS, LBS, 0, stream>>>(verts, center, longest, NV);
  }
  k_vkeys<<<(B_BATCH * NV + LBS - 1) / LBS, LBS, 0, stream>>>(verts, center, longest, vkeys, NV);

  auto sort_keys = [&](uint4* keys, int n) {
    int tiles = B_BATCH * (n / TLOC);
    k_bitonic_local_sort<<<tiles, LBS, 0, stream>>>(keys, n);
    int pairs = B_BATCH * (n / 2);
    int pblocks = (pairs + 255) / 256;
    for (int k = TLOC * 2; k <= n; k <<= 1) {
      for (int j = k >> 1; j >= TLOC; j >>= 1) {
        k_bitonic_global<<<pblocks, 256, 0, stream>>>(keys, n, k, j);
      }
      k_bitonic_local_merge<<<tiles, LBS, 0, stream>>>(keys, n, k);
    }
  };

  sort_keys(vkeys, NV);                                                   // vertex lexsort
  k_fkeys<<<(B_BATCH * NF + LBS - 1) / LBS, LBS, 0, stream>>>(faces, vkeys, fkeys, NV, NF);
  sort_keys(fkeys, NF);                                                   // face centroid lexsort
  k_emit<<<(B_BATCH * NF + LBS - 1) / LBS, LBS, 0, stream>>>(faces, vkeys, fkeys, out, NV, NF);
  k_edges<<<1, 64, 0, stream>>>(out, NF);
}